// Attn_3075196583966
// MI455X (gfx1250) — compile-verified
//
#include <hip/hip_runtime.h>

#define Nq 256
#define Mk 1024
#define Dd 512

typedef float v2f __attribute__((ext_vector_type(2)));
typedef float v8f __attribute__((ext_vector_type(8)));

// gfx1250 has a hardware TRANS tanh. Prefer the builtin (compiler manages the
// TRANS data hazard); fall back to inline asm with a v_nop for the co-exec
// hazard window.
__device__ __forceinline__ float fast_tanh(float x) {
#if __has_builtin(__builtin_amdgcn_tanhf)
  return __builtin_amdgcn_tanhf(x);
#elif __has_builtin(__builtin_amdgcn_tanh_f32)
  return __builtin_amdgcn_tanh_f32(x);
#else
  float r;
  asm volatile("v_tanh_f32 %0, %1\n\tv_nop" : "=v"(r) : "v"(x));
  return r;
#endif
}

// C[r][c] = sum_e A[r][e] * W[c][e] + bias[c]   (torch Linear)
// One wave per block; each wave owns one 16x16 output tile, accumulating with
// native f32 WMMA (16x16x4), 128 WMMAs over the K=512 dimension.
// transposeOut=1 writes out[c][r] (ld = rows) so the score kernel can read
// kp with unit-stride in m.
__global__ __launch_bounds__(32) void proj_wmma_f32(
    const float* __restrict__ A, const float* __restrict__ W,
    const float* __restrict__ bias, float* __restrict__ out,
    int rows, int cols, int kdim, int transposeOut)
{
  const int lane = threadIdx.x & 31;
  const int tilesPerRow = cols >> 4;
  const int tile = blockIdx.x;
  const int rowTile = (tile / tilesPerRow) << 4;
  const int colTile = (tile % tilesPerRow) << 4;
  const int half = lane >> 4;     // 0: lanes 0-15, 1: lanes 16-31
  const int l15  = lane & 15;
  const int col  = colTile + l15; // N index carried by this lane (B/C/D frags)
  const int rowA = rowTile + l15; // M index carried by this lane (A frag)

  // C/D 16x16 f32 layout: VGPR r holds row (rowTile + r + half*8), col = lane&15.
  // Init accumulator with the bias (depends only on col -> uniform across VGPRs).
  float bval = bias[col];
  v8f c;
#pragma unroll
  for (int i = 0; i < 8; ++i) c[i] = bval;

  const float* __restrict__ Arow = A + (size_t)rowA * kdim;
  const float* __restrict__ Wrow = W + (size_t)col  * kdim;

  // A 16x4 f32 frag: lanes 0-15 hold K={e0,e0+1}, lanes 16-31 hold K={e0+2,e0+3}.
  // B 4x16 mirrors with N striped across lanes. B[k][col] = W[col][k].
  for (int e0 = 0; e0 < kdim; e0 += 4) {
    const int ka = e0 + (half << 1);
    v2f a, b;
    a.x = Arow[ka];     a.y = Arow[ka + 1];
    b.x = Wrow[ka];     b.y = Wrow[ka + 1];
    c = __builtin_amdgcn_wmma_f32_16x16x4_f32(false, a, false, b,
                                              (short)0, c, false, false);
  }

  if (!transposeOut) {
#pragma unroll
    for (int r = 0; r < 8; ++r) {
      const int m = rowTile + r + (half << 3);
      out[(size_t)m * cols + col] = c[r];
    }
  } else {
#pragma unroll
    for (int r = 0; r < 8; ++r) {
      const int m = rowTile + r + (half << 3);
      out[(size_t)col * rows + m] = c[r];
    }
  }
}

// One workgroup per query row n. 256 threads; thread t owns keys m = 4t..4t+3.
// Score loop is TRANS-bound: 4 v_tanh_f32 + 4 v_fmac per d, kpT read as a
// coalesced float4 per lane. Then masked softmax over M via LDS reductions,
// probabilities parked in LDS, and a coalesced context GEMV over vp.
__global__ __launch_bounds__(256) void attn_score_softmax_ctx(
    const float* __restrict__ qp,    // [N][D]
    const float* __restrict__ kpT,   // [D][M]  (transposed kp)
    const float* __restrict__ vp,    // [M][D]
    const int*   __restrict__ mask,  // [N][M]
    const float* __restrict__ wvec,  // [D]
    const float* __restrict__ bw,    // [1]
    float* __restrict__ out)         // [N][D]
{
  __shared__ float qs[Dd];
  __shared__ float ws[Dd];
  __shared__ float ps[Mk];
  __shared__ float red[256];

  const int n = blockIdx.x;
  const int t = threadIdx.x;

  for (int d = t; d < Dd; d += 256) {
    qs[d] = qp[(size_t)n * Dd + d];
    ws[d] = wvec[d];
  }
  __syncthreads();

  // ---- additive scores for keys 4t..4t+3 ----
  const float4* __restrict__ kp4 = (const float4*)kpT;
  float s0 = 0.f, s1 = 0.f, s2 = 0.f, s3 = 0.f;
#pragma unroll 4
  for (int d = 0; d < Dd; ++d) {
    const float  qd = qs[d];
    const float  wd = ws[d];
    const float4 kk = kp4[(size_t)d * (Mk / 4) + t];
    s0 = fmaf(wd, fast_tanh(qd + kk.x), s0);
    s1 = fmaf(wd, fast_tanh(qd + kk.y), s1);
    s2 = fmaf(wd, fast_tanh(qd + kk.z), s2);
    s3 = fmaf(wd, fast_tanh(qd + kk.w), s3);
  }

  const float bwv   = bw[0];
  const int   mbase = t << 2;
  const int* __restrict__ mrow = mask + (size_t)n * Mk;
  s0 = (mrow[mbase + 0] == 1) ? (s0 + bwv) : -1.0e6f;
  s1 = (mrow[mbase + 1] == 1) ? (s1 + bwv) : -1.0e6f;
  s2 = (mrow[mbase + 2] == 1) ? (s2 + bwv) : -1.0e6f;
  s3 = (mrow[mbase + 3] == 1) ? (s3 + bwv) : -1.0e6f;

  // ---- softmax over M=1024: max reduction ----
  red[t] = fmaxf(fmaxf(s0, s1), fmaxf(s2, s3));
  __syncthreads();
  for (int off = 128; off > 0; off >>= 1) {
    if (t < off) red[t] = fmaxf(red[t], red[t + off]);
    __syncthreads();
  }
  const float mx = red[0];
  __syncthreads();

  const float e0 = __expf(s0 - mx);
  const float e1 = __expf(s1 - mx);
  const float e2 = __expf(s2 - mx);
  const float e3 = __expf(s3 - mx);

  red[t] = e0 + e1 + e2 + e3;
  __syncthreads();
  for (int off = 128; off > 0; off >>= 1) {
    if (t < off) red[t] += red[t + off];
    __syncthreads();
  }
  const float inv = 1.0f / red[0];
  __syncthreads();

  ps[mbase + 0] = e0 * inv;
  ps[mbase + 1] = e1 * inv;
  ps[mbase + 2] = e2 * inv;
  ps[mbase + 3] = e3 * inv;
  __syncthreads();

  // ---- context: out[n][d] = sum_m p[m] * vp[m][d] ----
  for (int d = t; d < Dd; d += 256) {
    float acc = 0.f;
#pragma unroll 4
    for (int m = 0; m < Mk; ++m)
      acc = fmaf(ps[m], vp[(size_t)m * Dd + d], acc);
    out[(size_t)n * Dd + d] = acc;
  }
}

extern "C" void kernel_launch(void* const* d_in, const int* in_sizes, int n_in,
                              void* d_out, int out_size, void* d_ws, size_t ws_size,
                              hipStream_t stream)
{
  const float* q    = (const float*)d_in[0];
  const float* k    = (const float*)d_in[1];
  const float* v    = (const float*)d_in[2];
  const int*   mask = (const int*)  d_in[3];
  const float* WQ   = (const float*)d_in[4];
  const float* bQ   = (const float*)d_in[5];
  const float* WK   = (const float*)d_in[6];
  const float* bK   = (const float*)d_in[7];
  const float* WV   = (const float*)d_in[8];
  const float* bV   = (const float*)d_in[9];
  const float* Ww   = (const float*)d_in[10];
  const float* bw   = (const float*)d_in[11];
  float* out = (float*)d_out;

  // workspace: qp [N*D] | kpT [D*M] | vp [M*D]  = 4.5 MB
  float* qp  = (float*)d_ws;
  float* kpT = qp  + (size_t)Nq * Dd;
  float* vp  = kpT + (size_t)Dd * Mk;

  const dim3 wblk(32);
  proj_wmma_f32<<<dim3((Nq / 16) * (Dd / 16)), wblk, 0, stream>>>(q, WQ, bQ, qp,  Nq, Dd, Dd, 0);
  proj_wmma_f32<<<dim3((Mk / 16) * (Dd / 16)), wblk, 0, stream>>>(k, WK, bK, kpT, Mk, Dd, Dd, 1);
  proj_wmma_f32<<<dim3((Mk / 16) * (Dd / 16)), wblk, 0, stream>>>(v, WV, bV, vp,  Mk, Dd, Dd, 0);

  attn_score_softmax_ctx<<<dim3(Nq), dim3(256), 0, stream>>>(qp, kpT, vp, mask, Ww, bw, out);
}